// Bridge_57329223467265
// MI455X (gfx1250) — compile-verified
//
#include <hip/hip_runtime.h>
#include <hip/hip_bf16.h>

typedef float v2f __attribute__((ext_vector_type(2)));
typedef float v4f __attribute__((ext_vector_type(4)));
typedef float v8f __attribute__((ext_vector_type(8)));

#define BDIM 4
#define SDIM 2048
#define DDIM 512
#define NCLS 16
#define LMAX (SDIM * (NCLS - 1))   // 30720
#define WPAD (DDIM + 2)            // LDS pad to break bank-conflict pathology

// ---------------------------------------------------------------------------
// Kernel 1: logits = x @ W + b via V_WMMA_F32_16X16X4_F32, then per-token
// argmax over 16 classes -> counts[B*S]. One wave per 16-token tile.
// ---------------------------------------------------------------------------
__global__ __launch_bounds__(256) void k_counts(const float* __restrict__ x,
                                                const float* __restrict__ W,
                                                const float* __restrict__ bias,
                                                int* __restrict__ counts) {
  __shared__ float sW[NCLS * WPAD];            // W transposed: [n][k]
  const int tid = threadIdx.x;
  for (int i = tid; i < DDIM * NCLS; i += 256) {
    const int k = i >> 4, n = i & 15;          // W is [D,16] row-major
    sW[n * WPAD + k] = W[i];
  }
  __syncthreads();

  const int lane  = tid & 31;
  const int wave  = (blockIdx.x * 256 + tid) >> 5;
  const int tile0 = wave << 4;                 // first token of this tile
  const int half  = lane >> 4;                 // 0 -> K{0,1}, 1 -> K{2,3}
  const int mn    = lane & 15;                 // A row / B column / class id
  const float* px = x + (size_t)(tile0 + mn) * DDIM + half * 2;
  const float* pw = sW + mn * WPAD + half * 2;

  v8f c = {0.f, 0.f, 0.f, 0.f, 0.f, 0.f, 0.f, 0.f};
#pragma unroll 4
  for (int k = 0; k < DDIM; k += 4) {
    v2f a  = *(const v2f*)(px + k);            // A: 16x4 f32 fragment
    v2f bw = *(const v2f*)(pw + k);            // B: 4x16 f32 fragment
    c = __builtin_amdgcn_wmma_f32_16x16x4_f32(false, a, false, bw,
                                              (short)0, c, false, false);
  }

  // add bias (per class = per lane%16), then butterfly argmax inside each
  // 16-lane half; ties resolve to the lowest class index (jnp.argmax rule).
  const float bn = bias[mn];
  float v[8];
  int   ci[8];
#pragma unroll
  for (int r = 0; r < 8; ++r) { v[r] = c[r] + bn; ci[r] = mn; }
#pragma unroll
  for (int m = 8; m >= 1; m >>= 1) {
#pragma unroll
    for (int r = 0; r < 8; ++r) {
      const float ov = __shfl_xor(v[r], m, 32);
      const int   oi = __shfl_xor(ci[r], m, 32);
      if (ov > v[r] || (ov == v[r] && oi < ci[r])) { v[r] = ov; ci[r] = oi; }
    }
  }
  if (mn == 0) {                               // lanes 0 and 16 write results
    const int base = tile0 + half * 8;         // D layout: vgpr r -> M=r(+8)
#pragma unroll
    for (int r = 0; r < 8; ++r) counts[base + r] = ci[r];
  }
}

// ---------------------------------------------------------------------------
// Kernel 2: inclusive cumsum of counts along S, one block per batch row.
// ---------------------------------------------------------------------------
__global__ __launch_bounds__(512) void k_cumsum(const int* __restrict__ counts,
                                                int* __restrict__ csum) {
  __shared__ int part[512];
  const int b = blockIdx.x;
  const int t = threadIdx.x;
  const int* in  = counts + b * SDIM;
  int*       out = csum   + b * SDIM;
  const int base = t * 4;
  const int v0 = in[base], v1 = in[base + 1], v2 = in[base + 2], v3 = in[base + 3];
  const int s1 = v0 + v1, s2 = s1 + v2, s3 = s2 + v3;
  part[t] = s3;
  __syncthreads();
  for (int off = 1; off < 512; off <<= 1) {    // Hillis-Steele over partials
    const int add = (t >= off) ? part[t - off] : 0;
    __syncthreads();
    part[t] += add;
    __syncthreads();
  }
  const int prefix = (t == 0) ? 0 : part[t - 1];
  out[base]     = prefix + v0;
  out[base + 1] = prefix + s1;
  out[base + 2] = prefix + s2;
  out[base + 3] = prefix + s3;
}

// ---------------------------------------------------------------------------
// Kernel 3: expand. One wave per output row (b, pos): binary-search the
// LDS-staged cumsum row, then stream-copy 512 floats (128 x b128) with
// non-temporal stores so the 252 MB output stream doesn't evict x from L2.
// ---------------------------------------------------------------------------
__global__ __launch_bounds__(256) void k_expand(const float* __restrict__ x,
                                                const int* __restrict__ csum,
                                                float* __restrict__ out) {
  __shared__ int sc[SDIM];
  const int b   = blockIdx.y;
  const int tid = threadIdx.x;
  for (int i = tid; i < SDIM; i += 256) sc[i] = csum[b * SDIM + i];
  __syncthreads();

  const int total = sc[SDIM - 1];
  const int lane  = tid & 31;
  const int pos   = blockIdx.x * 8 + (tid >> 5);
  v4f* dst = (v4f*)(out + ((size_t)b * LMAX + pos) * DDIM);

  if (pos >= total) {
    const v4f z = {0.f, 0.f, 0.f, 0.f};
#pragma unroll
    for (int j = 0; j < 4; ++j)
      __builtin_nontemporal_store(z, dst + lane + (j << 5));
  } else {
    int lo = 0, hi = SDIM;                     // searchsorted(csum, pos, right)
    while (lo < hi) {
      const int mid = (lo + hi) >> 1;
      if (sc[mid] <= pos) lo = mid + 1; else hi = mid;
    }
    const int idx = lo < SDIM ? lo : SDIM - 1;
    const v4f* src = (const v4f*)(x + ((size_t)b * SDIM + idx) * DDIM);
#pragma unroll
    for (int j = 0; j < 4; ++j)
      __builtin_nontemporal_store(src[lane + (j << 5)], dst + lane + (j << 5));
  }
}

// ---------------------------------------------------------------------------
extern "C" void kernel_launch(void* const* d_in, const int* in_sizes, int n_in,
                              void* d_out, int out_size, void* d_ws, size_t ws_size,
                              hipStream_t stream) {
  (void)in_sizes; (void)n_in; (void)out_size; (void)ws_size;
  const float* x    = (const float*)d_in[0];   // [B,S,D] f32
  const float* W    = (const float*)d_in[1];   // [D,16]  f32
  const float* bias = (const float*)d_in[2];   // [16]    f32
  float*       out  = (float*)d_out;           // [B,Lmax,D] f32

  int* counts = (int*)d_ws;                    // B*S ints
  int* csum   = counts + BDIM * SDIM;          // B*S ints (64 KB total ws)

  // 8192 tokens / (8 waves * 16 tokens) = 64 blocks; EXEC all-ones everywhere.
  k_counts<<<(BDIM * SDIM) / 128, 256, 0, stream>>>(x, W, bias, counts);
  k_cumsum<<<BDIM, 512, 0, stream>>>(counts, csum);
  dim3 g(LMAX / 8, BDIM);
  k_expand<<<g, 256, 0, stream>>>(x, csum, out);
}